// NEG_loss_88158498718050
// MI455X (gfx1250) — compile-verified
//
#include <hip/hip_runtime.h>

// NEG sampling loss, MI455X / gfx1250.
// One wave32 per batch row:
//   phase 1: 57 x GLOBAL_LOAD_ASYNC_TO_LDS_B128 gathers (512B each) straight
//            into a bank-conflict-free LDS tile -- no VGPR staging, all 57
//            random HBM reads in flight at once, single s_wait_asynccnt.
//   phase 2: 4 chained V_WMMA_F32_16X16X4_F32 tiles (K=128 -> 32 steps)
//            compute all 55 needed dot products as a 51x6 slice of
//            A(51x128, w-scaled) x B(128x6).
// Workload is gather-bandwidth bound (~470MB random reads, HBM 23.3 TB/s
// floor ~20us); WMMA does the dot-product core exactly in f32.

#define BATCH     16384
#define WLBL      5
#define NSAMP     5
#define EMBED     128
#define NVEC      57        // 1 diag_w + 1 in + 5 out + 25 neg_u + 25 neg_v
#define ROWSTRIDE 132       // 128 + 4 floats pad: (132*r + k) % 64 == (4r+k) % 64

typedef float v2f __attribute__((ext_vector_type(2)));
typedef float v8f __attribute__((ext_vector_type(8)));

__device__ __forceinline__ float lsig(float x) {
  // numerically stable log(sigmoid(x)) = min(x,0) - log1p(exp(-|x|))
  return fminf(x, 0.0f) - log1pf(__expf(-fabsf(x)));
}

__global__ __launch_bounds__(32)
void neg_loss_rows(const float* __restrict__ in_embed,
                   const float* __restrict__ out_embed,
                   const float* __restrict__ diag_w,
                   const int*   __restrict__ input_types,
                   const int*   __restrict__ input_ids,
                   const int*   __restrict__ out_labels,
                   const int*   __restrict__ noise_u,
                   const int*   __restrict__ cp_noise_v,
                   float*       __restrict__ row_out) {
  __shared__ float tile[NVEC * ROWSTRIDE];
  __shared__ int   s_idx[NVEC];
  __shared__ int   s_sel[NVEC];

  const int b    = blockIdx.x;
  const int lane = threadIdx.x;
  const int tp   = input_types[b];

  // ---- phase 0: build the gather table (row -> table, element index) ----
  for (int r = lane; r < NVEC; r += 32) {
    int sel, idx;
    if (r == 0)      { sel = 2; idx = tp; }                                       // diag_w[tp]
    else if (r == 1) { sel = 0; idx = input_ids[b]; }                             // in vec
    else if (r < 7)  { sel = 1; idx = out_labels[b * WLBL + (r - 2)]; }           // out vecs
    else if (r < 32) { sel = 0; idx = noise_u   [(tp * BATCH + b) * (WLBL*NSAMP) + (r - 7)]; }
    else             { sel = 1; idx = cp_noise_v[(tp * BATCH + b) * (WLBL*NSAMP) + (r - 32)]; }
    s_idx[r] = idx; s_sel[r] = sel;
  }
  __syncthreads();

  // ---- phase 1: async gather, global -> LDS, no VGPR staging ----
  // Each instruction moves 32 lanes x 16B = one 512B embedding vector.
  for (int r = 0; r < NVEC; ++r) {
    const int    sel = s_sel[r];
    const size_t idx = (size_t)s_idx[r];
    const float* src = (sel == 0) ? in_embed : ((sel == 1) ? out_embed : diag_w);
    const float* gaddr = src + idx * EMBED + lane * 4;
    const unsigned lds_off =
        (unsigned)(unsigned long long)(&tile[r * ROWSTRIDE + lane * 4]);
    asm volatile("global_load_async_to_lds_b128 %0, %1, off"
                 :: "v"(lds_off), "v"(gaddr)
                 : "memory");
  }
  asm volatile("s_wait_asynccnt 0x0" ::: "memory");
  __syncthreads();

  // ---- phase 1b: L2-reg terms from the LDS tile ----
  float regacc = 0.0f;
  for (int r = 0; r < NVEC; ++r) {
    const float4 v = *(const float4*)&tile[r * ROWSTRIDE + lane * 4];
    const float c = (r == 1) ? 5.0f : 1.0f;   // input row counted W=5 times
    regacc += c * (v.x*v.x + v.y*v.y + v.z*v.z + v.w*v.w);
  }

  // ---- phase 2: 4 WMMA tiles, A rows = w .* {in, neg_u[0..24], neg_v[0..24]},
  //               B cols = {out_0..out_4, in, 0...} ----
  const int hi = lane >> 4;
  const int nl = lane & 15;
  const int arow0 = (nl == 0) ? 1 : (6 + nl);          // in, neg_u[0..14]
  const int arow1 = 22 + nl;                           // neg_u[15..24], neg_v[0..5]
  const int arow2 = 38 + nl;                           // neg_v[6..21]
  const int arow3 = (nl < 3) ? (54 + nl) : 56;         // neg_v[22..24], dummies
  const int brow  = (nl == 5) ? 1 : (2 + nl);          // out_n (n<5), in (n==5)
  const float bmask = (nl < 6) ? 1.0f : 0.0f;

  v8f acc0 = {}, acc1 = {}, acc2 = {}, acc3 = {};
  for (int k0 = 0; k0 < EMBED; k0 += 4) {
    const int ko = k0 + 2 * hi;                        // per-half K offset
    const v2f w2 = *(const v2f*)&tile[ko];             // row 0 = diag_w (broadcast)
    const v2f bf = (*(const v2f*)&tile[brow * ROWSTRIDE + ko]) * bmask;
    v2f a;
    a = (*(const v2f*)&tile[arow0 * ROWSTRIDE + ko]) * w2;
    acc0 = __builtin_amdgcn_wmma_f32_16x16x4_f32(false, a, false, bf, (short)0, acc0, false, false);
    a = (*(const v2f*)&tile[arow1 * ROWSTRIDE + ko]) * w2;
    acc1 = __builtin_amdgcn_wmma_f32_16x16x4_f32(false, a, false, bf, (short)0, acc1, false, false);
    a = (*(const v2f*)&tile[arow2 * ROWSTRIDE + ko]) * w2;
    acc2 = __builtin_amdgcn_wmma_f32_16x16x4_f32(false, a, false, bf, (short)0, acc2, false, false);
    a = (*(const v2f*)&tile[arow3 * ROWSTRIDE + ko]) * w2;
    acc3 = __builtin_amdgcn_wmma_f32_16x16x4_f32(false, a, false, bf, (short)0, acc3, false, false);
  }

  // ---- phase 3: pick the 55 live entries out of the D tiles ----
  // D[m][n] lives at vgpr g = m%8, lane half = (m>=8), n = lane&15.
  float ssum = 0.0f;
  #pragma unroll
  for (int g = 0; g < 8; ++g) {
    const int m = g + 8 * hi;
    { // tile0: m==0 -> positives vs out_n (x2); m=1..15 -> neg_u[m-1] vs out_{(m-1)/5}
      const float d = acc0[g];
      if (m == 0) { if (nl < 5) ssum += 2.0f * lsig(d); }
      else if (nl == (m - 1) / NSAMP) ssum += lsig(-d);
    }
    { // tile1: m<10 -> neg_u[15+m] vs out_{(15+m)/5}; m>=10 -> neg_v[m-10] vs in (col 5)
      const float d = acc1[g];
      if (m < 10) { if (nl == (15 + m) / NSAMP) ssum += lsig(-d); }
      else if (nl == 5) ssum += lsig(-d);
    }
    // tile2: neg_v[6+m] vs in
    if (nl == 5) ssum += lsig(-acc2[g]);
    // tile3: m<3 -> neg_v[22+m] vs in
    if (m < 3 && nl == 5) ssum += lsig(-acc3[g]);
  }

  // per_row = (2*sum log_target + sum log_neg_u + sum log_neg_v) - reg
  float val = ssum - regacc;
  #pragma unroll
  for (int off = 16; off > 0; off >>= 1)
    val += __shfl_xor(val, off, 32);
  if (lane == 0) row_out[b] = val;
}

// Deterministic final reduction: fixed-order strided sums + fixed LDS tree.
__global__ __launch_bounds__(256)
void neg_loss_reduce(const float* __restrict__ row_out, float* __restrict__ out) {
  __shared__ float red[256];
  const int t = threadIdx.x;
  float s = 0.0f;
  for (int i = t; i < BATCH; i += 256) s += row_out[i];
  red[t] = s;
  __syncthreads();
  for (int st = 128; st > 0; st >>= 1) {
    if (t < st) red[t] += red[t + st];
    __syncthreads();
  }
  if (t == 0) out[0] = -red[0] / (2.0f * (float)BATCH);
}

extern "C" void kernel_launch(void* const* d_in, const int* in_sizes, int n_in,
                              void* d_out, int out_size, void* d_ws, size_t ws_size,
                              hipStream_t stream) {
  const float* in_embed   = (const float*)d_in[0];
  const float* out_embed  = (const float*)d_in[1];
  const float* diag_w     = (const float*)d_in[2];
  const int*   input_types= (const int*)d_in[3];
  const int*   input_ids  = (const int*)d_in[4];
  const int*   out_labels = (const int*)d_in[5];
  const int*   noise_u    = (const int*)d_in[6];
  const int*   cp_noise_v = (const int*)d_in[7];
  float*       out        = (float*)d_out;
  float*       ws         = (float*)d_ws;   // needs BATCH*4 = 64 KB

  neg_loss_rows<<<BATCH, 32, 0, stream>>>(in_embed, out_embed, diag_w,
                                          input_types, input_ids, out_labels,
                                          noise_u, cp_noise_v, ws);
  neg_loss_reduce<<<1, 256, 0, stream>>>(ws, out);
}